// KoopmanLQR_32255204393355
// MI455X (gfx1250) — compile-verified
//
#include <hip/hip_runtime.h>

#define BATCH  2048
#define KDIM   32
#define UDIM   8
#define UPAD   16
#define TSTEPS 200

typedef float v2f __attribute__((ext_vector_type(2)));
typedef float v8f __attribute__((ext_vector_type(8)));

// D = A(16x4 f32) * B(4x16 f32) + C, fp32 matrix pipe (V_WMMA_F32_16X16X4_F32)
static __device__ __forceinline__ v8f wmma4(v2f a, v2f b, v8f c) {
  return __builtin_amdgcn_wmma_f32_16x16x4_f32(false, a, false, b, (short)0, c, false, false);
}

// ---- WMMA operand staging from row-major LDS tiles ----
// A operand 16x4: VGPR0 holds K={0 | 2} split across lane halves, VGPR1 K={1 | 3}
static __device__ __forceinline__ v2f ldA(const float* M, int ld, int r0, int c0, int lane) {
  int m = lane & 15, kk = (lane >> 4) << 1;
  v2f a; a.x = M[(r0 + m) * ld + c0 + kk]; a.y = M[(r0 + m) * ld + c0 + kk + 1]; return a;
}
// A operand built from M^T (element (m,k) = M[k][m]) — free transpose via indexing
static __device__ __forceinline__ v2f ldAT(const float* M, int ld, int r0, int c0, int lane) {
  int m = lane & 15, kk = (lane >> 4) << 1;
  v2f a; a.x = M[(c0 + kk) * ld + r0 + m]; a.y = M[(c0 + kk + 1) * ld + r0 + m]; return a;
}
// B operand 4x16: rows striped across lanes, K-range half-split across lane halves
static __device__ __forceinline__ v2f ldB(const float* M, int ld, int r0, int c0, int lane) {
  int n = lane & 15, kk = (lane >> 4) << 1;
  v2f b; b.x = M[(r0 + kk) * ld + c0 + n]; b.y = M[(r0 + kk + 1) * ld + c0 + n]; return b;
}
// C/D 16x16: VGPR i -> M=i (lanes 0-15) / M=8+i (lanes 16-31), N=lane&15
static __device__ __forceinline__ v8f ldC(const float* M, int ld, int r0, int c0, int lane) {
  int n = lane & 15, mh = (lane >> 4) << 3;
  v8f c;
#pragma unroll
  for (int i = 0; i < 8; ++i) c[i] = M[(r0 + mh + i) * ld + c0 + n];
  return c;
}
static __device__ __forceinline__ void stC(float* M, int ld, int r0, int c0, int lane, v8f c) {
  int n = lane & 15, mh = (lane >> 4) << 3;
#pragma unroll
  for (int i = 0; i < 8; ++i) M[(r0 + mh + i) * ld + c0 + n] = c[i];
}

__global__ __launch_bounds__(32) void koopman_lqr_riccati(
    const float* __restrict__ gA, const float* __restrict__ gB,
    const float* __restrict__ gQ, const float* __restrict__ gR,
    const float* __restrict__ gGoals,
    float* __restrict__ outK, float* __restrict__ outk)
{
  __shared__ float sA[KDIM * KDIM];
  __shared__ float sQ[KDIM * KDIM];
  __shared__ float sV[KDIM * KDIM];
  __shared__ float sM1[KDIM * KDIM];
  __shared__ float sABK[KDIM * KDIM];
  __shared__ float sB[KDIM * UPAD];      // B zero-padded to 32x16
  __shared__ float sBtV[UPAD * KDIM];
  __shared__ float sT1[UPAD * KDIM];
  __shared__ float sKg[UPAD * KDIM];
  __shared__ float sR[UPAD * UPAD];      // R zero-padded to 16x16
  __shared__ float sVuu[UPAD * UPAD];    // Vuu then padded inverse
  __shared__ float sGJ[UDIM * 2 * UDIM]; // 8x16 augmented [Vuu | I]
  __shared__ float sv[KDIM];
  __shared__ float sg[KDIM];
  __shared__ float sw[UDIM];

  const int b = blockIdx.x;
  const int lane = threadIdx.x;

  const float* Ab = gA + (size_t)b * KDIM * KDIM;
  const float* Bb = gB + (size_t)b * KDIM * UDIM;
  const float* Qb = gQ + (size_t)b * KDIM * KDIM;
  const float* Rb = gR + (size_t)b * UDIM * UDIM;
  const float* Gb = gGoals + (size_t)b * (TSTEPS + 1) * KDIM;

  // Stage per-batch constants into LDS (lane = row)
#pragma unroll 4
  for (int j = 0; j < KDIM; ++j) {
    float q = Qb[lane * KDIM + j];
    sA[lane * KDIM + j] = Ab[lane * KDIM + j];
    sQ[lane * KDIM + j] = q;
    sV[lane * KDIM + j] = q;               // V0 = Q
  }
#pragma unroll
  for (int u = 0; u < UPAD; ++u)
    sB[lane * UPAD + u] = (u < UDIM) ? Bb[lane * UDIM + u] : 0.0f;
#pragma unroll
  for (int t = 0; t < 8; ++t) {
    int idx = lane * 8 + t, r = idx >> 4, c = idx & 15;
    sR[idx] = (r < UDIM && c < UDIM) ? Rb[r * UDIM + c] : 0.0f;
  }
  sg[lane] = Gb[TSTEPS * KDIM + lane];     // goals[:, -1, :]
  __syncthreads();
  {                                        // v0 = Q * goal_T (Q symmetric: column read, bank-conflict free)
    float a0 = 0.f;
#pragma unroll 4
    for (int m = 0; m < KDIM; ++m) a0 += sQ[m * KDIM + lane] * sg[m];
    sv[lane] = a0;
  }
  __syncthreads();

  // ---- Hoist loop-invariant WMMA operands into registers (~120 VGPRs) ----
  v2f rBt[8];       // B^T as A-operand        (step 1)
  v2f rBb[8];       // B   as B-operand        (step 2)
  v2f rBa[2][4];    // B   as A-operand tiles  (step 7)
  v2f rAb[2][8];    // A   as B-operand tiles  (step 4b): [tj][kk]
  v2f rAt[2][8];    // A^T as A-operand tiles  (step 8):  [ti][kk]
  v8f rR;           // R   C-layout            (step 2)
#pragma unroll
  for (int k4 = 0; k4 < 8; ++k4) {
    rBt[k4] = ldAT(sB, UPAD, 0, 4 * k4, lane);
    rBb[k4] = ldB(sB, UPAD, 4 * k4, 0, lane);
  }
#pragma unroll
  for (int t = 0; t < 2; ++t) {
#pragma unroll
    for (int k4 = 0; k4 < 8; ++k4) {
      rAb[t][k4] = ldB(sA, KDIM, 4 * k4, 16 * t, lane);
      rAt[t][k4] = ldAT(sA, KDIM, 16 * t, 4 * k4, lane);
    }
#pragma unroll
    for (int k4 = 0; k4 < 4; ++k4) rBa[t][k4] = ldA(sB, UPAD, 16 * t, 4 * k4, lane);
  }
  rR = ldC(sR, UPAD, 0, 0, lane);

#pragma unroll 1
  for (int i = TSTEPS - 1; i >= 0; --i) {
    if (i > 0) __builtin_prefetch(&Gb[(i - 1) * KDIM], 0, 0);  // global_prefetch_b8
    sg[lane] = Gb[i * KDIM + lane];

    // Load V's B-operand tiles once; shared by steps (1) and (8)
    v2f rV[2][8];
#pragma unroll
    for (int tj = 0; tj < 2; ++tj)
#pragma unroll
      for (int k4 = 0; k4 < 8; ++k4)
        rV[tj][k4] = ldB(sV, KDIM, 4 * k4, 16 * tj, lane);

    // (1) BtV = B^T * V   (16x32, rows 8..15 zero via padding)
#pragma unroll
    for (int tj = 0; tj < 2; ++tj) {
      v8f acc = {};
#pragma unroll
      for (int k4 = 0; k4 < 8; ++k4)
        acc = wmma4(rBt[k4], rV[tj][k4], acc);
      stC(sBtV, KDIM, 0, 16 * tj, lane, acc);
    }
    // (8, hoisted) M1 = A^T * V — reuses rV tiles
#pragma unroll
    for (int ti = 0; ti < 2; ++ti) {
#pragma unroll
      for (int tj = 0; tj < 2; ++tj) {
        v8f acc = {};
#pragma unroll
        for (int k4 = 0; k4 < 8; ++k4)
          acc = wmma4(rAt[ti][k4], rV[tj][k4], acc);
        stC(sM1, KDIM, 16 * ti, 16 * tj, lane, acc);
      }
    }
    __syncthreads();

    // (2) Vuu = BtV*B + R  -> sVuu (padded 16x16)
    {
      v8f acc = rR;
#pragma unroll
      for (int k4 = 0; k4 < 8; ++k4)
        acc = wmma4(ldA(sBtV, KDIM, 0, 4 * k4, lane), rBb[k4], acc);
      stC(sVuu, UPAD, 0, 0, lane, acc);
    }
    __syncthreads();

    // (3) 8x8 inverse: lane-parallel Gauss-Jordan, 4 elements/lane over all 32 lanes
#pragma unroll
    for (int t = 0; t < 4; ++t) {
      int idx = t * 32 + lane, r = idx >> 4, c = idx & 15;
      sGJ[idx] = (c < UDIM) ? sVuu[r * UPAD + c] : ((c - UDIM) == r ? 1.0f : 0.0f);
    }
    __syncthreads();
#pragma unroll 1
    for (int p = 0; p < UDIM; ++p) {
      float inv = 1.0f / sGJ[p * 16 + p];           // broadcast load
      if (lane < 16) sGJ[p * 16 + lane] *= inv;     // scale pivot row
      __syncthreads();
      float fv[4], pc[4], ov[4];
      int rr[4];
#pragma unroll
      for (int t = 0; t < 4; ++t) {
        int idx = t * 32 + lane; rr[t] = idx >> 4; int c = idx & 15;
        ov[t] = sGJ[idx];
        fv[t] = sGJ[rr[t] * 16 + p];
        pc[t] = sGJ[p * 16 + c];
      }
      __syncthreads();
#pragma unroll
      for (int t = 0; t < 4; ++t) {
        int idx = t * 32 + lane;
        if (rr[t] != p) sGJ[idx] = ov[t] - fv[t] * pc[t];
      }
      __syncthreads();
    }
#pragma unroll
    for (int t = 0; t < 8; ++t) {          // padded inverse -> sVuu
      int idx = t * 32 + lane, r = idx >> 4, c = idx & 15;
      sVuu[idx] = (r < UDIM && c < UDIM) ? sGJ[r * 16 + UDIM + c] : 0.0f;
    }
    __syncthreads();

    // (4) T1 = Vuu_inv * BtV ; Kg = T1 * A
#pragma unroll
    for (int tj = 0; tj < 2; ++tj) {
      v8f acc = {};
#pragma unroll
      for (int k4 = 0; k4 < 4; ++k4)
        acc = wmma4(ldA(sVuu, UPAD, 0, 4 * k4, lane), ldB(sBtV, KDIM, 4 * k4, 16 * tj, lane), acc);
      stC(sT1, KDIM, 0, 16 * tj, lane, acc);
    }
    __syncthreads();
#pragma unroll
    for (int tj = 0; tj < 2; ++tj) {
      v8f acc = {};
#pragma unroll
      for (int k4 = 0; k4 < 8; ++k4)
        acc = wmma4(ldA(sT1, KDIM, 0, 4 * k4, lane), rAb[tj][k4], acc);
      stC(sKg, KDIM, 0, 16 * tj, lane, acc);
    }
    __syncthreads();

    // (5) kf = Vuu_inv * (B^T v)
    if (lane < UDIM) {
      float s = 0.f;
#pragma unroll 4
      for (int k = 0; k < KDIM; ++k) s += sB[k * UPAD + lane] * sv[k];
      sw[lane] = s;
    }
    __syncthreads();
    if (lane < UDIM) {
      float s = 0.f;
#pragma unroll
      for (int u2 = 0; u2 < UDIM; ++u2) s += sVuu[lane * UPAD + u2] * sw[u2];
      outk[((size_t)i * BATCH + b) * UDIM + lane] = s;
    }

    // (6) store K gains [i][b][8][32]: 2x b128 per lane, coalesced
    {
      const size_t kb = ((size_t)i * BATCH + b) * (UDIM * KDIM);
      const float4* kg4 = (const float4*)sKg;   // first 8 rows = 64 float4s
      float4* o4 = (float4*)(outK + kb);
      o4[lane * 2]     = kg4[lane * 2];
      o4[lane * 2 + 1] = kg4[lane * 2 + 1];
    }

    // (7) A_BK = A - B*Kg
#pragma unroll
    for (int ti = 0; ti < 2; ++ti) {
#pragma unroll
      for (int tj = 0; tj < 2; ++tj) {
        v8f acc = {};
#pragma unroll
        for (int k4 = 0; k4 < 4; ++k4)
          acc = wmma4(rBa[ti][k4], ldB(sKg, KDIM, 4 * k4, 16 * tj, lane), acc);
        v8f at = ldC(sA, KDIM, 16 * ti, 16 * tj, lane);
#pragma unroll
        for (int e = 0; e < 8; ++e) at[e] -= acc[e];
        stC(sABK, KDIM, 16 * ti, 16 * tj, lane, at);
      }
    }
    __syncthreads();

    // (9) V = M1 * A_BK + Q
#pragma unroll
    for (int ti = 0; ti < 2; ++ti) {
#pragma unroll
      for (int tj = 0; tj < 2; ++tj) {
        v8f acc = ldC(sQ, KDIM, 16 * ti, 16 * tj, lane);
#pragma unroll
        for (int k4 = 0; k4 < 8; ++k4)
          acc = wmma4(ldA(sM1, KDIM, 16 * ti, 4 * k4, lane), ldB(sABK, KDIM, 4 * k4, 16 * tj, lane), acc);
        stC(sV, KDIM, 16 * ti, 16 * tj, lane, acc);
      }
    }

    // (10) v = A_BK^T v + Q * goal_i   (Q symmetric: column reads, conflict-free)
    {
      float vn = 0.f;
#pragma unroll 4
      for (int k = 0; k < KDIM; ++k) vn += sABK[k * KDIM + lane] * sv[k];
#pragma unroll 4
      for (int m = 0; m < KDIM; ++m) vn += sQ[m * KDIM + lane] * sg[m];
      __syncthreads();
      sv[lane] = vn;
    }
    __syncthreads();
  }
}

extern "C" void kernel_launch(void* const* d_in, const int* in_sizes, int n_in,
                              void* d_out, int out_size, void* d_ws, size_t ws_size,
                              hipStream_t stream) {
  (void)in_sizes; (void)n_in; (void)out_size; (void)d_ws; (void)ws_size;
  const float* A     = (const float*)d_in[0];
  const float* B     = (const float*)d_in[1];
  const float* Q     = (const float*)d_in[2];
  const float* R     = (const float*)d_in[3];
  const float* goals = (const float*)d_in[4];
  float* outK = (float*)d_out;
  float* outk = outK + (size_t)TSTEPS * BATCH * UDIM * KDIM;  // k_ff after K_gains
  hipLaunchKernelGGL(koopman_lqr_riccati, dim3(BATCH), dim3(32), 0, stream,
                     A, B, Q, R, goals, outK, outk);
}